// freebase_bidirection_hnode_prompt_layer_feature_weighted_sum_21534966022309
// MI455X (gfx1250) — compile-verified
//
#include <hip/hip_runtime.h>

#define WTILE 32          // edges staged per wave-tile (1 dword per lane per array)
constexpr int kD = 64;    // feature dim

// ---- gfx1250 async global->LDS copy (ASYNCcnt-tracked) -------------------
// ISA: GLOBAL_LOAD_ASYNC_TO_LDS_B32  vdst(LDS addr VGPR), vaddr(64b), off
__device__ __forceinline__ void async_cp_b32(const void* g, void* l) {
#if defined(__AMDGCN__)
  unsigned loff =
      (unsigned)(unsigned long long)(__attribute__((address_space(3))) void*)l;
  asm volatile("global_load_async_to_lds_b32 %0, %1, off"
               :: "v"(loff), "v"(g) : "memory");
#endif
}

template <int N>
__device__ __forceinline__ void wait_async() {
#if defined(__AMDGCN__)
  asm volatile("s_wait_asynccnt %0" :: "i"(N) : "memory");
#endif
}

// ---- kernel 1: zero the output (harness poisons it) ----------------------
__global__ void zero_out_kernel(float4* __restrict__ out, int n4) {
  int i = blockIdx.x * blockDim.x + threadIdx.x;
  if (i < n4) out[i] = make_float4(0.f, 0.f, 0.f, 0.f);
}

// ---- kernel 2: emb = elu(ge * w), float4 vectorized ----------------------
__global__ void emb_elu_kernel(const float* __restrict__ ge,
                               const float* __restrict__ w,
                               float* __restrict__ emb, int n4) {
  int i = blockIdx.x * blockDim.x + threadIdx.x;
  if (i >= n4) return;
  int idx = i * 4;
  float4 g = *(const float4*)(ge + idx);
  float4 wv = *(const float4*)(w + (idx & (kD - 1)));   // row length 64, idx%64
  float4 r; float x;
  x = g.x * wv.x; r.x = x > 0.f ? x : expm1f(x);
  x = g.y * wv.y; r.y = x > 0.f ? x : expm1f(x);
  x = g.z * wv.z; r.z = x > 0.f ? x : expm1f(x);
  x = g.w * wv.w; r.w = x > 0.f ? x : expm1f(x);
  *(float4*)(emb + idx) = r;
}

// ---- kernel 3: wave-autonomous edge scatter, async double-buffered -------
// tiles [0, tilesPerSet)             -> graph       (coef = etype<5 ? 2 : 1)
// tiles [tilesPerSet, 2*tilesPerSet) -> trans_graph (coef = etype in {6,14,30})
// Each wave stages its own 32-edge index tile into a private LDS slice and
// waits only on its own ASYNCcnt -> zero workgroup barriers.
__global__ void edge_scatter_kernel(const float* __restrict__ emb,
                                    const int* __restrict__ efeat,
                                    const int* __restrict__ gsrc,
                                    const int* __restrict__ gdst,
                                    const int* __restrict__ tsrc,
                                    const int* __restrict__ tdst,
                                    float* __restrict__ out,
                                    int E, int tilesPerSet) {
  __shared__ int sh[8][2][3][WTILE];        // [wave][buf][etype/src/dst][edge]
  const int lane = threadIdx.x & 31;
  const int wave = threadIdx.x >> 5;        // 0..7 (wave32)
  const int nwavesTotal = (int)gridDim.x * (int)(blockDim.x >> 5);
  const int waveId = (int)blockIdx.x * (int)(blockDim.x >> 5) + wave;
  const int totalTiles = 2 * tilesPerSet;

  auto stage = [&](int tile, int buf) {
    int set = tile >= tilesPerSet;
    int tIn = set ? tile - tilesPerSet : tile;
    int e   = tIn * WTILE + lane;
    int ec  = e < E ? e : E - 1;            // clamp: every lane a valid address
    const int* sp = set ? tsrc : gsrc;
    const int* dp = set ? tdst : gdst;
    async_cp_b32(efeat + ec, &sh[wave][buf][0][lane]);
    async_cp_b32(sp + ec,    &sh[wave][buf][1][lane]);
    async_cp_b32(dp + ec,    &sh[wave][buf][2][lane]);
  };

  int t = waveId;
  if (t < totalTiles) stage(t, 0);
  int it = 0;
  for (; t < totalTiles; t += nwavesTotal, ++it) {
    int buf = it & 1;
    int tn  = t + nwavesTotal;
    if (tn < totalTiles) {                  // prefetch next tile, then wait for
      stage(tn, buf ^ 1);                   // this wave's current-tile loads
      wait_async<3>();                      // (async loads complete in order)
    } else {
      wait_async<0>();
    }

    int set  = t >= tilesPerSet;
    int base = (set ? t - tilesPerSet : t) * WTILE;
    for (int k = 0; k < WTILE; ++k) {       // one edge per wave per step
      if (base + k >= E) break;
      // broadcast reads; force into SGPRs so gather/atomic use saddr form
      int et = __builtin_amdgcn_readfirstlane(sh[wave][buf][0][k]);
      float coef;
      if (!set) {
        coef = (et < 5) ? 2.0f : 1.0f;
      } else {
        if (!(et == 6 || et == 14 || et == 30)) continue;  // skip 29/32 edges
        coef = 1.0f;
      }
      int s = __builtin_amdgcn_readfirstlane(sh[wave][buf][1][k]);
      int d = __builtin_amdgcn_readfirstlane(sh[wave][buf][2][k]);
      // 32 lanes x float2 = one 64-float row: one coalesced 256B gather
      const float2 v = *(const float2*)(emb + (size_t)s * kD + lane * 2);
      float* o = out + (size_t)d * kD + lane * 2;
      atomicAdd(o,     v.x * coef);         // L2-resident f32 atomics (no-rtn)
      atomicAdd(o + 1, v.y * coef);
    }
  }
}

extern "C" void kernel_launch(void* const* d_in, const int* in_sizes, int n_in,
                              void* d_out, int out_size, void* d_ws, size_t ws_size,
                              hipStream_t stream) {
  const float* ge = (const float*)d_in[0];   // [N, 64]
  const float* w  = (const float*)d_in[1];   // [1, 64]
  const int*   ef = (const int*)d_in[2];     // [E]
  const int*   gs = (const int*)d_in[3];     // [E]
  const int*   gd = (const int*)d_in[4];     // [E]
  const int*   ts = (const int*)d_in[5];     // [E]
  const int*   td = (const int*)d_in[6];     // [E]
  float* out = (float*)d_out;                // [N, 64]
  float* emb = (float*)d_ws;                 // [N, 64] scratch

  const int ND = in_sizes[0];                // N * 64
  const int E  = in_sizes[2];

  const int tb = 256;
  const int n4 = ND / 4;
  zero_out_kernel<<<(n4 + tb - 1) / tb, tb, 0, stream>>>((float4*)out, n4);
  emb_elu_kernel<<<(n4 + tb - 1) / tb, tb, 0, stream>>>(ge, w, emb, n4);

  const int tilesPerSet = (E + WTILE - 1) / WTILE;
  const int totalTiles  = 2 * tilesPerSet;
  const int wavesPerBlock = tb / 32;         // 8
  int blocks = (totalTiles + wavesPerBlock - 1) / wavesPerBlock;
  if (blocks > 2048) blocks = 2048;          // ~16K waves, ~4 tiles each
  edge_scatter_kernel<<<blocks, tb, 0, stream>>>(emb, ef, gs, gd, ts, td, out,
                                                 E, tilesPerSet);
}